// Conv1dLayer_13365938225847
// MI455X (gfx1250) — compile-verified
//
#include <hip/hip_runtime.h>

typedef __attribute__((ext_vector_type(2))) float v2f;
typedef __attribute__((ext_vector_type(8))) float v8f;

#define B_      1024
#define CIN     64
#define COUT    64
#define W_      15
#define HW      225          // 15*15
#define PADH    17           // padded spatial edge
#define PITCHC  66           // channel pitch (floats) in LDS pixel-major layout
#define LDS_FLOATS (PADH*PADH*PITCHC)   // 19074 floats = 76.3 KB

// One block per (batch image, direction). 256 threads = 8 waves (wave32).
// LDS layout: x_pad[hp][wp][ci], hp/wp in [0,17), ci pitch 66 (8B-aligned v2f
// reads, ~conflict-free banks). Halo rows/cols zeroed -> padding is implicit.
__global__ __launch_bounds__(256) void dirconv_wmma_f32(
    const float* __restrict__ x0, const float* __restrict__ x1,
    const float* __restrict__ x2, const float* __restrict__ x3,
    const float* __restrict__ wgt,   // (3, COUT, CIN)
    const float* __restrict__ bias,  // (COUT)
    float* __restrict__ out)         // 4 x (B, COUT, 15, 15) concatenated
{
    __shared__ float lds[LDS_FLOATS];
    const int b   = blockIdx.x;
    const int d   = blockIdx.y;
    const int tid = threadIdx.x;

    const float* xsel[4] = {x0, x1, x2, x3};
    const float* x = xsel[d] + (size_t)b * (CIN * HW);

    // ---- stage image into LDS (zero halo, then transposed interior fill) ----
    for (int i = tid; i < LDS_FLOATS; i += 256) lds[i] = 0.0f;
    __syncthreads();
    for (int i = tid; i < CIN * HW; i += 256) {
        int ci = i / HW;
        int p  = i - ci * HW;
        int h  = p / W_;
        int w  = p - h * W_;
        lds[((h + 1) * PADH + (w + 1)) * PITCHC + ci] = x[i];
    }
    __syncthreads();

    // ---- per-direction tap shifts (scalar, uniform per block) ----
    // d0: horizontal, d1: vertical, d2: main diag, d3: anti-diag
    const int DYt[4][3] = {{0,0,0}, {-1,0,1}, {-1,0,1}, {1,0,-1}};
    const int DXt[4][3] = {{-1,0,1}, {0,0,0}, {-1,0,1}, {-1,0,1}};
    int ddel[3];
#pragma unroll
    for (int t = 0; t < 3; ++t)
        ddel[t] = (DYt[d][t] * PADH + DXt[d][t]) * PITCHC;

    const int wv     = tid >> 5;
    const int lane   = tid & 31;
    const int lanelo = lane & 15;
    const int hi     = lane >> 4;   // 0: K/ci 0..1, 1: K/ci 2..3 half
    const int ct     = wv & 3;      // output-channel tile (4 x 16)
    const int halfw  = wv >> 2;     // pixel-tile phase (2-way split)

    // ---- register-cache A fragments (weights), K = t*64 + ci ----
    // A(16x4) per lane: M = lanelo, K pair (2*hi, 2*hi+1) -> v2f contiguous in ci
    v2f af[48];
#pragma unroll
    for (int t = 0; t < 3; ++t) {
#pragma unroll
        for (int k = 0; k < 16; ++k) {
            int c = 4 * k + 2 * hi;
            af[t * 16 + k] =
                *(const v2f*)(wgt + ((t * COUT + ct * 16 + lanelo) * CIN + c));
        }
    }

    float bv[8];
#pragma unroll
    for (int r = 0; r < 8; ++r) bv[r] = bias[ct * 16 + r + 8 * hi];

    float* ob = out + (size_t)(d * B_ + b) * (COUT * HW);

    // ---- 15 pixel tiles of 16 (last one masked at store) ----
    for (int pt = halfw; pt < 15; pt += 2) {
        int pix = pt * 16 + lanelo;
        int pv  = pix < HW ? pix : HW - 1;      // clamp keeps LDS reads in range
        int h   = pv / W_;
        int w   = pv - h * W_;
        int pbase = ((h + 1) * PADH + (w + 1)) * PITCHC + 2 * hi;

        v8f acc;
#pragma unroll
        for (int r = 0; r < 8; ++r) acc[r] = bv[r];

#pragma unroll
        for (int t = 0; t < 3; ++t) {
            int tb = pbase + ddel[t];
#pragma unroll
            for (int k = 0; k < 16; ++k) {
                // B(4x16) per lane: N = lanelo's pixel, K pair (2*hi, 2*hi+1)
                v2f bf = *(const v2f*)(&lds[tb + 4 * k]);
                acc = __builtin_amdgcn_wmma_f32_16x16x4_f32(
                    false, af[t * 16 + k], false, bf,
                    (short)0, acc, false, false);
            }
        }

        if (pix < HW) {
            // D layout: vgpr r, lanes 0-15 -> co = r, lanes 16-31 -> co = r+8
#pragma unroll
            for (int r = 0; r < 8; ++r)
                ob[(ct * 16 + r + 8 * hi) * HW + pv] = acc[r];
        }
    }
}

extern "C" void kernel_launch(void* const* d_in, const int* in_sizes, int n_in,
                              void* d_out, int out_size, void* d_ws, size_t ws_size,
                              hipStream_t stream) {
    const float* x0   = (const float*)d_in[0];
    const float* x1   = (const float*)d_in[1];
    const float* x2   = (const float*)d_in[2];
    const float* x3   = (const float*)d_in[3];
    const float* wgt  = (const float*)d_in[4];
    const float* bias = (const float*)d_in[5];
    float* out = (float*)d_out;

    dim3 grid(B_, 4, 1);
    dim3 block(256, 1, 1);
    hipLaunchKernelGGL(dirconv_wmma_f32, grid, block, 0, stream,
                       x0, x1, x2, x3, wgt, bias, out);
}